// GCNNet_80857054314690
// MI455X (gfx1250) — compile-verified
//
#include <hip/hip_runtime.h>

// ---------------------------------------------------------------------------
// GCN (3x GCNConv + linear head) for MI455X / gfx1250.
// fp32 end-to-end; GEMMs on V_WMMA_F32_16X16X4_F32; aggregation is pull-style
// over a device-built CSR (no atomics in the 128-wide hot path).
// ---------------------------------------------------------------------------

typedef __attribute__((ext_vector_type(2))) float v2f;
typedef __attribute__((ext_vector_type(8))) float v8f;

#define DH   128   // D_IN == D_HID == D_EMB
#define DCAT 384   // 2*D_HID + D_EMB
#define DLAB 10

// ---------------- small utility kernels ----------------

__global__ void k_zero_i32(int* __restrict__ p, int n) {
  int i = blockIdx.x * blockDim.x + threadIdx.x;
  if (i < n) p[i] = 0;
}

// in-degree histogram over target nodes (col)
__global__ void k_count(const int* __restrict__ col, int* __restrict__ counts, int e) {
  int i = blockIdx.x * blockDim.x + threadIdx.x;
  if (i < e) atomicAdd(&counts[col[i]], 1);
}

// dinv = rsqrt(deg), deg = in-degree + 1 (self loop) -> always > 0
__global__ void k_dinv(const int* __restrict__ counts, float* __restrict__ dinv, int n) {
  int i = blockIdx.x * blockDim.x + threadIdx.x;
  if (i < n) dinv[i] = rsqrtf((float)(counts[i] + 1));
}

// ---------------- exclusive prefix scan (3 stages) ----------------

__global__ void k_scan_block(const int* __restrict__ counts, int* __restrict__ excl,
                             int* __restrict__ bsums, int n) {
  __shared__ int sh[256];
  int tid = threadIdx.x, gid = blockIdx.x * 256 + tid;
  int v = (gid < n) ? counts[gid] : 0;
  sh[tid] = v;
  __syncthreads();
  for (int off = 1; off < 256; off <<= 1) {
    int t = (tid >= off) ? sh[tid - off] : 0;
    __syncthreads();
    sh[tid] += t;
    __syncthreads();
  }
  if (gid < n) excl[gid] = sh[tid] - v;           // exclusive within block
  if (tid == 255) bsums[blockIdx.x] = sh[255];    // block total
}

__global__ void k_scan_sums(int* __restrict__ bsums, int nb) {  // nb <= 1024
  __shared__ int sh[1024];
  int tid = threadIdx.x;
  int v = (tid < nb) ? bsums[tid] : 0;
  sh[tid] = v;
  __syncthreads();
  for (int off = 1; off < 1024; off <<= 1) {
    int t = (tid >= off) ? sh[tid - off] : 0;
    __syncthreads();
    sh[tid] += t;
    __syncthreads();
  }
  if (tid < nb) bsums[tid] = sh[tid] - v;         // exclusive block offsets
}

__global__ void k_scan_add(int* __restrict__ offs, const int* __restrict__ bsums,
                           int* __restrict__ cursor, int n, int e) {
  int gid = blockIdx.x * 256 + threadIdx.x;
  if (gid < n) {
    int o = offs[gid] + bsums[blockIdx.x];
    offs[gid] = o;
    cursor[gid] = o;
  }
  if (gid == 0) offs[n] = e;
}

// CSR fill: bucket edges by target; store source id + symmetric-norm weight
__global__ void k_csr_fill(const int* __restrict__ ei, const float* __restrict__ dinv,
                           int* __restrict__ cursor, int* __restrict__ csr_src,
                           float* __restrict__ csr_w, int e) {
  int i = blockIdx.x * blockDim.x + threadIdx.x;
  if (i < e) {
    int r = ei[i];          // source (row)
    int c = ei[e + i];      // target (col)
    int pos = atomicAdd(&cursor[c], 1);
    csr_src[pos] = r;
    csr_w[pos] = dinv[r] * dinv[c];
  }
}

// ---------------- fp32 WMMA GEMM: y[N x 128] = x[N x K=128] @ W[128 x 128] ----------------
// one wave per 16x16 output tile; K-loop of 32 x V_WMMA_F32_16X16X4_F32

__global__ void k_gemm128(const float* __restrict__ x, int xstride,
                          const float* __restrict__ W,
                          float* __restrict__ y, int n) {
  const int tm = blockIdx.x * 16;
  const int tn = blockIdx.y * 16;
  const int ln = threadIdx.x;
  const int half = ln >> 4;       // 0: K even pair, 1: K odd pair
  const int lan  = ln & 15;

  int m = tm + lan;
  if (m >= n) m = n - 1;          // clamp loads; stores are guarded
  const float* xr = x + (long)m * xstride;

  v8f c = {};
#pragma unroll
  for (int k = 0; k < DH; k += 4) {
    const int ka = k + 2 * half;
    // A 16x4: lanes 0-15 hold K={k,k+1}, lanes 16-31 hold K={k+2,k+3}, M = lane&15
    v2f a;
    a.x = xr[ka];
    a.y = xr[ka + 1];
    // B 4x16: lanes 0-15 hold K={k,k+1}, lanes 16-31 hold K={k+2,k+3}, N = lane&15
    v2f b;
    b.x = W[(long)ka * DH + tn + lan];
    b.y = W[(long)(ka + 1) * DH + tn + lan];
    c = __builtin_amdgcn_wmma_f32_16x16x4_f32(false, a, false, b, (short)0, c, false, false);
  }

  const int m0 = 8 * half;        // C: VGPR r -> M = r (lanes 0-15) / 8+r (lanes 16-31)
#pragma unroll
  for (int r = 0; r < 8; ++r) {
    int row = tm + m0 + r;
    if (row < n) y[(long)row * DH + tn + lan] = c[r];
  }
}

// ---------------- pull aggregation: h[i] = relu?( sum_j w*xw[src] + dinv^2*xw[i] + b ) ----------------
// one block (128 threads) per node, thread = feature lane; gathers hit L2 (xw = 51MB << 192MB)

__global__ void k_aggregate(const float* __restrict__ xw, const float* __restrict__ dinv,
                            const int* __restrict__ offs, const int* __restrict__ src,
                            const float* __restrict__ w, const float* __restrict__ bias,
                            float* __restrict__ h, int hstride, int relu, int n) {
  const int i = blockIdx.x;
  if (i >= n) return;
  const int f = threadIdx.x;

  const float di = dinv[i];
  float acc = xw[(long)i * DH + f] * di * di;    // self-loop term

  const int e0 = offs[i];
  const int e1 = offs[i + 1];
  for (int j = e0; j < e1; ++j) {
    const int s = src[j];
    if (j + 1 < e1)
      __builtin_prefetch(&xw[(long)src[j + 1] * DH + f], 0, 0);  // global_prefetch_b8
    acc += xw[(long)s * DH + f] * w[j];
  }

  float v = acc + bias[f];
  h[(long)i * hstride + f] = relu ? fmaxf(v, 0.0f) : v;
}

// ---------------- projection: out[N x 10] = hcat[N x 384] @ Wp[384 x 10] + bp ----------------
// WMMA tile padded to 16 columns; lanes with n >= 10 carry zeros.

__global__ void k_proj(const float* __restrict__ hcat, const float* __restrict__ Wp,
                       const float* __restrict__ bp, float* __restrict__ out, int n) {
  const int tm = blockIdx.x * 16;
  const int ln = threadIdx.x;
  const int half = ln >> 4;
  const int lan  = ln & 15;

  int m = tm + lan;
  if (m >= n) m = n - 1;
  const float* xr = hcat + (long)m * DCAT;

  v8f c = {};
#pragma unroll 8
  for (int k = 0; k < DCAT; k += 4) {
    const int ka = k + 2 * half;
    v2f a;
    a.x = xr[ka];
    a.y = xr[ka + 1];
    v2f b;
    b.x = (lan < DLAB) ? Wp[(long)ka * DLAB + lan] : 0.0f;
    b.y = (lan < DLAB) ? Wp[(long)(ka + 1) * DLAB + lan] : 0.0f;
    c = __builtin_amdgcn_wmma_f32_16x16x4_f32(false, a, false, b, (short)0, c, false, false);
  }

  if (lan < DLAB) {
    const float bb = bp[lan];
    const int m0 = 8 * half;
#pragma unroll
    for (int r = 0; r < 8; ++r) {
      int row = tm + m0 + r;
      if (row < n) out[(long)row * DLAB + lan] = c[r] + bb;
    }
  }
}

// ---------------- host orchestration ----------------

extern "C" void kernel_launch(void* const* d_in, const int* in_sizes, int n_in,
                              void* d_out, int out_size, void* d_ws, size_t ws_size,
                              hipStream_t stream) {
  const float* feat = (const float*)d_in[0];
  const int*   ei   = (const int*)d_in[1];     // [2, E]: rows then cols
  const float* W1 = (const float*)d_in[2];
  const float* b1 = (const float*)d_in[3];
  const float* W2 = (const float*)d_in[4];
  const float* b2 = (const float*)d_in[5];
  const float* W3 = (const float*)d_in[6];
  const float* b3 = (const float*)d_in[7];
  const float* Wp = (const float*)d_in[8];
  const float* bp = (const float*)d_in[9];

  const int N = in_sizes[0] / DH;
  const int E = in_sizes[1] / 2;
  float* out = (float*)d_out;

  // workspace carve-out (256B aligned)
  char* p = (char*)d_ws;
  auto take = [&](size_t bytes) -> char* {
    char* q = p;
    p += (bytes + 255) & ~(size_t)255;
    return q;
  };
  int*   counts  = (int*)  take((size_t)N * 4);         // reused as cursor after scan
  int*   offs    = (int*)  take(((size_t)N + 1) * 4);
  float* dinv    = (float*)take((size_t)N * 4);
  int*   bsums   = (int*)  take(1024 * 4);
  int*   csr_src = (int*)  take((size_t)E * 4);
  float* csr_w   = (float*)take((size_t)E * 4);
  float* xw      = (float*)take((size_t)N * DH * 4);
  float* hcat    = (float*)take((size_t)N * DCAT * 4);  // [h1 | h2 | h3] rows

  const int nb = (N + 255) / 256;   // <= 1024 for N <= 262144
  const int eb = (E + 255) / 256;

  // ---- graph normalization + CSR build (once, reused by all 3 layers) ----
  k_zero_i32 <<<nb, 256, 0, stream>>>(counts, N);
  k_count    <<<eb, 256, 0, stream>>>(ei + E, counts, E);
  k_dinv     <<<nb, 256, 0, stream>>>(counts, dinv, N);
  k_scan_block<<<nb, 256, 0, stream>>>(counts, offs, bsums, N);
  k_scan_sums <<<1, 1024, 0, stream>>>(bsums, nb);
  k_scan_add  <<<nb, 256, 0, stream>>>(offs, bsums, counts /*cursor*/, N, E);
  k_csr_fill  <<<eb, 256, 0, stream>>>(ei, dinv, counts, csr_src, csr_w, E);

  // ---- 3 GCN layers, writing straight into the concatenated buffer ----
  const int tilesM = (N + 15) / 16;
  dim3 gg(tilesM, DH / 16);

  k_gemm128  <<<gg, 32, 0, stream>>>(feat, DH, W1, xw, N);
  k_aggregate<<<N, DH, 0, stream>>>(xw, dinv, offs, csr_src, csr_w, b1, hcat + 0,      DCAT, 1, N);

  k_gemm128  <<<gg, 32, 0, stream>>>(hcat + 0, DCAT, W2, xw, N);
  k_aggregate<<<N, DH, 0, stream>>>(xw, dinv, offs, csr_src, csr_w, b2, hcat + DH,     DCAT, 1, N);

  k_gemm128  <<<gg, 32, 0, stream>>>(hcat + DH, DCAT, W3, xw, N);
  k_aggregate<<<N, DH, 0, stream>>>(xw, dinv, offs, csr_src, csr_w, b3, hcat + 2 * DH, DCAT, 0, N);

  // ---- head ----
  k_proj<<<tilesM, 32, 0, stream>>>(hcat, Wp, bp, out, N);
}